// ComplexMul2dParallel_58591943852264
// MI455X (gfx1250) — compile-verified
//
#include <hip/hip_runtime.h>
#include <stdint.h>

typedef __attribute__((ext_vector_type(2))) float v2f;
typedef __attribute__((ext_vector_type(8))) float v8f;

#define BATCH 32
#define CIN   128
#define COUT  128
#define MODES 4160   // 64 * 65
#define MT    8      // modes per workgroup (== number of waves)
#define OC    32     // Cout slice per workgroup
#define KC    16     // complex-K chunk staged in LDS
#define AROW  20     // A row stride (floats): 8 complex interleaved (16) + 4 pad
#define BROW  36     // B row stride (floats): Bneg plane (16) + Bswap plane (16) + 4 pad
#define ALDS  (KC * BATCH * AROW)   // 10240 floats (40 KB)
#define BLDS  (KC * OC * BROW)      // 18432 floats (72 KB)

__device__ __forceinline__ uint32_t lds_off(const void* p) {
  // Generic LDS address: addr[31:0] is the LDS offset (ISA 10.2 aperture rules).
  return (uint32_t)(uintptr_t)p;
}

__device__ __forceinline__ void async_b128(uint32_t lds_dst, const void* gsrc) {
  // GLOBAL_LOAD_ASYNC_TO_LDS_B128: per-lane 16B global->LDS, tracked by ASYNCcnt.
  asm volatile("global_load_async_to_lds_b128 %0, %1, off"
               :: "v"(lds_dst), "v"((uint64_t)(uintptr_t)gsrc)
               : "memory");
}

__device__ __forceinline__ void wait_async0() {
#if __has_builtin(__builtin_amdgcn_s_wait_asynccnt)
  __builtin_amdgcn_s_wait_asynccnt(0);
#else
  asm volatile("s_wait_asynccnt 0x0" ::: "memory");
#endif
}

__global__ __launch_bounds__(256) void ComplexMul2dParallel_58591943852264_kernel(
    const float* __restrict__ x,   // [B, Cin, 64, 65, 2]
    const float* __restrict__ w,   // [Cin, Cout, 64, 65, 2]
    float* __restrict__ out)       // [B, Cout, 64, 65, 2]
{
  __shared__ __align__(16) float smem[ALDS + BLDS];  // 112 KB
  float* ldsA = smem;            // [KC][BATCH][AROW] interleaved (re,im) x 8 modes
  float* ldsB = smem + ALDS;     // [KC][OC][BROW]  = {Bneg 16f, Bswap 16f, pad}

  const int tid  = threadIdx.x;
  const int wave = tid >> 5;   // 0..7 : this wave's mode within the chunk
  const int lane = tid & 31;
  const int lm   = lane & 15;  // lane within half -> M / N index
  const int lh   = lane >> 4;  // lane half -> which complex-K of the K'=4 step

  const int mode_base = blockIdx.x * MT;   // 520 chunks
  const int o_base    = blockIdx.y * OC;   // 4 slices

  v8f accr[2][2];
  v8f acci[2][2];
  const v8f zero = {0.f, 0.f, 0.f, 0.f, 0.f, 0.f, 0.f, 0.f};
  for (int mt = 0; mt < 2; ++mt)
    for (int nt = 0; nt < 2; ++nt) {
      accr[mt][nt] = zero;
      acci[mt][nt] = zero;
    }

  for (int kc = 0; kc < CIN / KC; ++kc) {
    const int kg0 = kc * KC;

    // ---- A chunk: async copy, verbatim interleaved. 4 x b128 per (k,b) row.
    for (int p = tid; p < KC * BATCH; p += 256) {
      const int k = p >> 5;
      const int b = p & 31;
      const float* gsrc = x + (((size_t)b * CIN + (kg0 + k)) * MODES + mode_base) * 2;
      const uint32_t ldst = lds_off(ldsA + ((size_t)k * BATCH + b) * AROW);
      async_b128(ldst +  0, gsrc +  0);
      async_b128(ldst + 16, gsrc +  4);
      async_b128(ldst + 32, gsrc +  8);
      async_b128(ldst + 48, gsrc + 12);
    }

    // ---- B chunk: load, build Bneg=(Br,-Bi) and Bswap=(Bi,Br) planes in LDS.
    for (int p = tid; p < KC * OC; p += 256) {
      const int k = p >> 5;
      const int o = p & 31;
      const float4* src =
          (const float4*)(w + (((size_t)(kg0 + k) * COUT + (o_base + o)) * MODES + mode_base) * 2);
      float4* dn = (float4*)(ldsB + ((size_t)k * OC + o) * BROW);        // Bneg
      float4* ds = dn + 4;                                               // Bswap (+64B)
#pragma unroll
      for (int q = 0; q < 4; ++q) {
        const float4 v = src[q];
        float4 pn, ps;
        pn.x = v.x;  pn.y = -v.y;  pn.z = v.z;  pn.w = -v.w;
        ps.x = v.y;  ps.y =  v.x;  ps.z = v.w;  ps.w =  v.z;
        dn[q] = pn;
        ds[q] = ps;
      }
    }

    wait_async0();
    __syncthreads();

    // ---- compute: real GEMM over K' = 2*KC; each K'4-step = 2 complex k.
    // Every operand is one aligned b64 from LDS -- no shuffles, no negates here.
#pragma unroll
    for (int s = 0; s < KC / 2; ++s) {
      const int kA = 2 * s + lh;  // complex-k row this lane-half reads

      v2f A2[2], Bn[2], Bs[2];
#pragma unroll
      for (int mt = 0; mt < 2; ++mt)
        A2[mt] = *(const v2f*)(ldsA + ((size_t)kA * BATCH + (mt * 16 + lm)) * AROW + wave * 2);
#pragma unroll
      for (int nt = 0; nt < 2; ++nt) {
        const float* base = ldsB + ((size_t)kA * OC + (nt * 16 + lm)) * BROW + wave * 2;
        Bn[nt] = *(const v2f*)(base);
        Bs[nt] = *(const v2f*)(base + 16);
      }

#pragma unroll
      for (int mt = 0; mt < 2; ++mt)
#pragma unroll
        for (int nt = 0; nt < 2; ++nt) {
          accr[mt][nt] = __builtin_amdgcn_wmma_f32_16x16x4_f32(
              false, A2[mt], false, Bn[nt], (short)0, accr[mt][nt], false, false);
          acci[mt][nt] = __builtin_amdgcn_wmma_f32_16x16x4_f32(
              false, A2[mt], false, Bs[nt], (short)0, acci[mt][nt], false, false);
        }
    }
    __syncthreads();
  }

  // ---- epilogue: restage through LDS as [b][o][mode] so global stores coalesce
  float* ldsO = smem;  // 32*32*8 complex = 64 KB, fits in the 112 KB region
#pragma unroll
  for (int mt = 0; mt < 2; ++mt)
#pragma unroll
    for (int nt = 0; nt < 2; ++nt)
#pragma unroll
      for (int r = 0; r < 8; ++r) {
        const int b = mt * 16 + r + 8 * lh;   // C/D layout: VGPR r, lane half -> M
        const int o = nt * 16 + lm;           // lane -> N
        float2 v;
        v.x = accr[mt][nt][r];
        v.y = acci[mt][nt][r];
        *(float2*)(ldsO + (((size_t)b * OC + o) * MT + wave) * 2) = v;
      }
  __syncthreads();

  for (int p = tid; p < BATCH * OC; p += 256) {
    const int b = p >> 5;
    const int o = p & 31;
    const float4* src = (const float4*)(ldsO + (size_t)p * MT * 2);
    float4* dst = (float4*)(out + (((size_t)b * COUT + (o_base + o)) * MODES + mode_base) * 2);
    dst[0] = src[0]; dst[1] = src[1]; dst[2] = src[2]; dst[3] = src[3];
  }
}

extern "C" void kernel_launch(void* const* d_in, const int* in_sizes, int n_in,
                              void* d_out, int out_size, void* d_ws, size_t ws_size,
                              hipStream_t stream) {
  (void)in_sizes; (void)n_in; (void)out_size; (void)d_ws; (void)ws_size;
  const float* x = (const float*)d_in[0];   // input  [32,128,64,65,2] f32
  const float* w = (const float*)d_in[1];   // weight [128,128,64,65,2] f32
  float* o = (float*)d_out;                 // output [32,128,64,65,2] f32

  dim3 grid(MODES / MT, COUT / OC);  // (520, 4)
  dim3 block(256);
  ComplexMul2dParallel_58591943852264_kernel<<<grid, block, 0, stream>>>(x, w, o);
}